// DiffJPEG_90134183674619
// MI455X (gfx1250) — compile-verified
//
#include <hip/hip_runtime.h>

// DiffJPEG fused single-pass kernel for MI455X (gfx1250, wave32).
// Memory-bound problem (~192MB traffic, 23.3TB/s -> ~8us floor), so: one fused
// pass, f32 end-to-end, 8x8 DCTs batched 4-at-a-time into 16x16 f32 WMMAs via
// a block-diagonal DCT matrix, transposes through padded LDS.

typedef float v2f __attribute__((ext_vector_type(2)));
typedef float v8f __attribute__((ext_vector_type(8)));

// 8-point DCT-II matrix, D[i][j] = alpha_i * cos((2j+1) i pi / 16)
__constant__ float DCTM[8][8] = {
  { 0.353553391f,  0.353553391f,  0.353553391f,  0.353553391f,  0.353553391f,  0.353553391f,  0.353553391f,  0.353553391f},
  { 0.490392640f,  0.415734806f,  0.277785117f,  0.097545161f, -0.097545161f, -0.277785117f, -0.415734806f, -0.490392640f},
  { 0.461939766f,  0.191341716f, -0.191341716f, -0.461939766f, -0.461939766f, -0.191341716f,  0.191341716f,  0.461939766f},
  { 0.415734806f, -0.097545161f, -0.490392640f, -0.277785117f,  0.277785117f,  0.490392640f,  0.097545161f, -0.415734806f},
  { 0.353553391f, -0.353553391f, -0.353553391f,  0.353553391f,  0.353553391f, -0.353553391f, -0.353553391f,  0.353553391f},
  { 0.277785117f, -0.490392640f,  0.097545161f,  0.415734806f, -0.415734806f, -0.097545161f,  0.490392640f, -0.277785117f},
  { 0.191341716f, -0.461939766f,  0.461939766f, -0.191341716f, -0.191341716f,  0.461939766f, -0.461939766f,  0.191341716f},
  { 0.097545161f, -0.277785117f,  0.415734806f, -0.490392640f,  0.490392640f, -0.415734806f,  0.277785117f, -0.097545161f}
};

// Block-diagonal 16x16 DCT matrix element: diag(D, D)
__device__ __forceinline__ float dbd(int m, int k) {
  return ((m >> 3) == (k >> 3)) ? DCTM[m & 7][k & 7] : 0.0f;
}

__device__ __forceinline__ v8f wmma4(v2f a, v2f b, v8f c) {
  // D = A(16x4) x B(4x16) + C, f32. 4 chained -> full 16x16x16 GEMM.
  return __builtin_amdgcn_wmma_f32_16x16x4_f32(false, a, false, b, (short)0, c,
                                               false, false);
}

// Computes rec = E * round((E*X*E^T)/q)*q * E^T for E = Dbd (fdct) then
// E = Dbd^T (idct), as two GEMM pairs of form  E * (E * X^T)^T.
// A operand: constant fragments aD/aDT. B operand: transposed reads (getX or
// LDS scratch with stride-17 padding for conflict-free column access).
template <typename GX>
__device__ __forceinline__ void dct_pipe(GX getX, const float* qt, float* scr,
                                         float* outp, int ostride,
                                         const v2f* aD, const v2f* aDT) {
  const int lane = threadIdx.x & 31;
  const int mn   = lane & 15;   // N (B/D cols) == M (A rows) for this lane
  const int hi   = lane >> 4;   // half-wave select

  v2f bf[4];
  v8f acc;

  // ---- GEMM1: S = Dbd * X^T        (B[k][n] = X[n][k]) ----
#pragma unroll
  for (int c = 0; c < 4; ++c) {
    const int k0 = 4 * c + 2 * hi;
    bf[c].x = getX(mn, k0);
    bf[c].y = getX(mn, k0 + 1);
  }
  acc = {};
#pragma unroll
  for (int c = 0; c < 4; ++c) acc = wmma4(aD[c], bf[c], acc);
#pragma unroll
  for (int r = 0; r < 8; ++r) scr[(r + 8 * hi) * 17 + mn] = acc[r];

  // ---- GEMM2: coef = Dbd * S^T ----
#pragma unroll
  for (int c = 0; c < 4; ++c) {
    const int k0 = 4 * c + 2 * hi;
    bf[c].x = scr[mn * 17 + k0];
    bf[c].y = scr[mn * 17 + k0 + 1];
  }
  acc = {};
#pragma unroll
  for (int c = 0; c < 4; ++c) acc = wmma4(aD[c], bf[c], acc);

  // ---- quantize / dequantize (each 8x8 quadrant is one block's DCT) ----
#pragma unroll
  for (int r = 0; r < 8; ++r) {
    const float q = qt[r * 8 + (mn & 7)];
    acc[r] = rintf(acc[r] / q) * q;   // rte round, matches jnp.round
  }
#pragma unroll
  for (int r = 0; r < 8; ++r) scr[(r + 8 * hi) * 17 + mn] = acc[r];

  // ---- GEMM3: U = Dbd^T * coefq^T ----
#pragma unroll
  for (int c = 0; c < 4; ++c) {
    const int k0 = 4 * c + 2 * hi;
    bf[c].x = scr[mn * 17 + k0];
    bf[c].y = scr[mn * 17 + k0 + 1];
  }
  acc = {};
#pragma unroll
  for (int c = 0; c < 4; ++c) acc = wmma4(aDT[c], bf[c], acc);
#pragma unroll
  for (int r = 0; r < 8; ++r) scr[(r + 8 * hi) * 17 + mn] = acc[r];

  // ---- GEMM4: rec = Dbd^T * U^T ----
#pragma unroll
  for (int c = 0; c < 4; ++c) {
    const int k0 = 4 * c + 2 * hi;
    bf[c].x = scr[mn * 17 + k0];
    bf[c].y = scr[mn * 17 + k0 + 1];
  }
  acc = {};
#pragma unroll
  for (int c = 0; c < 4; ++c) acc = wmma4(aDT[c], bf[c], acc);
#pragma unroll
  for (int r = 0; r < 8; ++r) outp[(r + 8 * hi) * ostride + mn] = acc[r];
}

__global__ __launch_bounds__(256) void diffjpeg_fused(
    const float* __restrict__ x, const float* __restrict__ yt,
    const float* __restrict__ ct, float* __restrict__ out) {
  __shared__ float yIn[32 * 33];       // Y - 128 (input to DCT)
  __shared__ float cbF[32 * 33];       // Cb - 128, full res
  __shared__ float crF[32 * 33];       // Cr - 128, full res
  __shared__ float yRec[32 * 33];      // reconstructed Y - 128
  __shared__ float cbRec[16 * 17];     // reconstructed Cb - 128 (subsampled)
  __shared__ float crRec[16 * 17];
  __shared__ float scr[6 * 16 * 17];   // per-wave transpose scratch
  __shared__ float qtY[64], qtC[64];

  const int tid = threadIdx.x;
  const int b   = blockIdx.z;
  const int H0  = blockIdx.y * 32;
  const int W0  = blockIdx.x * 32;
  const long ps = 1024L * 1024L;       // plane stride

  // quant tables: clip(round(t), 1, 32767)
  if (tid < 64)       qtY[tid]      = fminf(fmaxf(rintf(yt[tid]), 1.0f), 32767.0f);
  else if (tid < 128) qtC[tid - 64] = fminf(fmaxf(rintf(ct[tid - 64]), 1.0f), 32767.0f);

  // ---- load 32x32 RGB region (float4 / B128 coalesced), RGB->YCbCr ----
  {
    const int row = tid >> 3;
    const int col = (tid & 7) * 4;
    const long off = (long)(H0 + row) * 1024 + (W0 + col);
    const float4 rv = *(const float4*)(x + ((long)b * 3 + 0) * ps + off);
    const float4 gv = *(const float4*)(x + ((long)b * 3 + 1) * ps + off);
    const float4 bv = *(const float4*)(x + ((long)b * 3 + 2) * ps + off);
#pragma unroll
    for (int i = 0; i < 4; ++i) {
      const float r  = ((const float*)&rv)[i] * 255.0f;
      const float g  = ((const float*)&gv)[i] * 255.0f;
      const float bl = ((const float*)&bv)[i] * 255.0f;
      yIn[row * 33 + col + i] = 0.299f * r + 0.587f * g + 0.114f * bl - 128.0f;
      cbF[row * 33 + col + i] = -0.168736f * r - 0.331264f * g + 0.5f * bl;
      crF[row * 33 + col + i] = 0.5f * r - 0.418688f * g - 0.081312f * bl;
    }
  }
  __syncthreads();

  const int wave = tid >> 5;
  const int lane = tid & 31;
  const int mn   = lane & 15;
  const int hi   = lane >> 4;

  if (wave < 6) {
    // Constant A fragments of Dbd and Dbd^T (16x4 chunks, K-slices 0,4,8,12)
    v2f aD[4], aDT[4];
#pragma unroll
    for (int c = 0; c < 4; ++c) {
      const int k0 = 4 * c + 2 * hi;
      aD[c].x  = dbd(mn, k0);      aD[c].y  = dbd(mn, k0 + 1);
      aDT[c].x = dbd(k0, mn);      aDT[c].y = dbd(k0 + 1, mn);
    }
    float* ws = &scr[wave * 272];

    if (wave < 4) {  // four 16x16 Y tiles of the 32x32 region
      const int tr = (wave >> 1) * 16, tc = (wave & 1) * 16;
      const float* Xb = &yIn[tr * 33 + tc];
      auto gx = [=](int rr, int cc) { return Xb[rr * 33 + cc]; };
      dct_pipe(gx, qtY, ws, &yRec[tr * 33 + tc], 33, aD, aDT);
    } else {         // Cb (wave 4) / Cr (wave 5): 2x2-mean subsample on the fly
      const float* src = (wave == 4) ? cbF : crF;
      float* dst       = (wave == 4) ? cbRec : crRec;
      auto gx = [=](int rr, int cc) {
        const int r2 = rr * 2, c2 = cc * 2;
        return 0.25f * (src[r2 * 33 + c2] + src[r2 * 33 + c2 + 1] +
                        src[(r2 + 1) * 33 + c2] + src[(r2 + 1) * 33 + c2 + 1]);
      };
      dct_pipe(gx, qtC, ws, dst, 17, aD, aDT);
    }
  }
  __syncthreads();

  // ---- upsample chroma + YCbCr->RGB, float4 coalesced stores ----
  {
    const int row = tid >> 3;
    const int col = (tid & 7) * 4;
    float4 Rv, Gv, Bv;
#pragma unroll
    for (int i = 0; i < 4; ++i) {
      const int cc = col + i;
      const float Y  = yRec[row * 33 + cc] + 128.0f;
      const float cb = cbRec[(row >> 1) * 17 + (cc >> 1)];
      const float cr = crRec[(row >> 1) * 17 + (cc >> 1)];
      const float inv = 1.0f / 255.0f;
      ((float*)&Rv)[i] = (Y + 1.402f * cr) * inv;
      ((float*)&Gv)[i] = (Y - 0.344136f * cb - 0.714136f * cr) * inv;
      ((float*)&Bv)[i] = (Y + 1.772f * cb) * inv;
    }
    const long off = (long)(H0 + row) * 1024 + (W0 + col);
    *(float4*)(out + ((long)b * 3 + 0) * ps + off) = Rv;
    *(float4*)(out + ((long)b * 3 + 1) * ps + off) = Gv;
    *(float4*)(out + ((long)b * 3 + 2) * ps + off) = Bv;
  }
}

extern "C" void kernel_launch(void* const* d_in, const int* in_sizes, int n_in,
                              void* d_out, int out_size, void* d_ws,
                              size_t ws_size, hipStream_t stream) {
  (void)in_sizes; (void)n_in; (void)out_size; (void)d_ws; (void)ws_size;
  const float* xp = (const float*)d_in[0];
  const float* yt = (const float*)d_in[1];
  const float* ct = (const float*)d_in[2];
  float* outp = (float*)d_out;
  dim3 grid(1024 / 32, 1024 / 32, 8);   // (W tiles, H tiles, batch)
  diffjpeg_fused<<<grid, dim3(256), 0, stream>>>(xp, yt, ct, outp);
}